// GraphAttentionLayer_69544110457400
// MI455X (gfx1250) — compile-verified
//
#include <hip/hip_runtime.h>
#include <hip/hip_bf16.h>
#include <math.h>

typedef float v2f __attribute__((ext_vector_type(2)));
typedef float v8f __attribute__((ext_vector_type(8)));

#define NW 20000
#define NS 512
#define E_SW 160000
#define E_SS 262144
#define E_WS 160000
#define ENC_NEG_INF 0x007FFFFFu   // monotonic-uint encoding of -inf

// ---------------------------------------------------------------- utilities

__device__ __forceinline__ float blockReduceSum(float v, float* red) {
  int t = threadIdx.x;
  red[t] = v; __syncthreads();
  for (int w = 128; w > 0; w >>= 1) { if (t < w) red[t] += red[t + w]; __syncthreads(); }
  float r = red[0]; __syncthreads();
  return r;
}

__device__ __forceinline__ float blockReduceMax(float v, float* red) {
  int t = threadIdx.x;
  red[t] = v; __syncthreads();
  for (int w = 128; w > 0; w >>= 1) { if (t < w) red[t] = fmaxf(red[t], red[t + w]); __syncthreads(); }
  float r = red[0]; __syncthreads();
  return r;
}

__device__ __forceinline__ float sigmoidf(float x) { return 1.f / (1.f + __expf(-x)); }
__device__ __forceinline__ float lrelu(float x) { return x > 0.f ? x : 0.01f * x; }

__global__ void k_fill_f32(float* p, float v, int n) {
  int i = blockIdx.x * 256 + threadIdx.x;
  if (i < n) p[i] = v;
}
__global__ void k_fill_u32(unsigned* p, unsigned v, int n) {
  int i = blockIdx.x * 256 + threadIdx.x;
  if (i < n) p[i] = v;
}
__global__ void k_sigmoid(float* x, int n) {
  int i = blockIdx.x * 256 + threadIdx.x;
  if (i < n) x[i] = sigmoidf(x[i]);
}

// ------------------------------------------------ projected attention vecs
// (z @ a_half) == x @ (W @ a_half): never materialize z matrices.
__global__ void k_proj(const float* W_s, const float* W_w, const float* W_sup,
                       const float* a_sw, const float* a_ss, const float* a_ws,
                       const float* a_ssup, const float* a_wsup, float* proj) {
  int b = blockIdx.x, i = threadIdx.x;
  const float* Wm; const float* av;
  switch (b) {
    case 0: Wm = W_s;   av = a_sw;         break; // q1: sw src (sent)
    case 1: Wm = W_s;   av = a_ss;         break; // q2: ss src
    case 2: Wm = W_s;   av = a_ss + 256;   break; // q3: ss dst
    case 3: Wm = W_s;   av = a_ws + 256;   break; // q4: ws dst (sent)
    case 4: Wm = W_s;   av = a_ssup;       break; // q5: ssuper src
    case 5: Wm = W_w;   av = a_sw + 256;   break; // p1: sw dst (word)
    case 6: Wm = W_w;   av = a_ws;         break; // p2: ws src (word)
    case 7: Wm = W_w;   av = a_wsup;       break; // p3: wsuper src
    case 8: Wm = W_sup; av = a_ssup + 256; break; // r1: ssuper dst
    default: Wm = W_sup; av = a_wsup + 256; break;// r2: wsuper dst
  }
  float s = 0.f;
  for (int k = 0; k < 256; ++k) s += Wm[i * 256 + k] * av[k];
  proj[b * 256 + i] = s;
}

// -------------------------------------------- per-node logits + sh + dots
__global__ void k_sent(const float* sent_h, const float* score, const float* super_h,
                       const float* proj, float* sh, float* ls_sw, float* ls_ss,
                       float* ld_ss, float* ld_ws, float* ls_ssup, float* dots) {
  __shared__ float red[256];
  int s = blockIdx.x, t = threadIdx.x;
  float x = score[s] * sent_h[s * 256 + t];
  sh[s * 256 + t] = x;
  float r;
  r = blockReduceSum(x * proj[t], red);        if (t == 0) ls_sw[s] = r;
  r = blockReduceSum(x * proj[256 + t], red);  if (t == 0) ls_ss[s] = r;
  r = blockReduceSum(x * proj[512 + t], red);  if (t == 0) ld_ss[s] = r;
  r = blockReduceSum(x * proj[768 + t], red);  if (t == 0) ld_ws[s] = r;
  r = blockReduceSum(x * proj[1024 + t], red); if (t == 0) ls_ssup[s] = r;
  r = blockReduceSum(x * super_h[t], red);     if (t == 0) dots[s] = r;
}

__global__ void k_word(const float* word_h, const float* proj,
                       float* ld_sw, float* ls_ws, float* ls_wsup) {
  __shared__ float red[256];
  int w = blockIdx.x, t = threadIdx.x;
  float x = word_h[w * 256 + t];
  float r;
  r = blockReduceSum(x * proj[1280 + t], red); if (t == 0) ld_sw[w] = r;
  r = blockReduceSum(x * proj[1536 + t], red); if (t == 0) ls_ws[w] = r;
  r = blockReduceSum(x * proj[1792 + t], red); if (t == 0) ls_wsup[w] = r;
}

__global__ void k_sup_scalars(const float* super_h, const float* proj,
                              float* ld_ssup, float* ld_wsup) {
  __shared__ float red[256];
  int t = threadIdx.x;
  float x = super_h[t];
  float r;
  r = blockReduceSum(x * proj[2048 + t], red); if (t == 0) ld_ssup[0] = r;
  r = blockReduceSum(x * proj[2304 + t], red); if (t == 0) ld_wsup[0] = r;
}

// ------------------------------------------------------------- WMMA GEMM
// C[M,256] = sigmoid?(cat(A0,A1,A2)[M,Ktot] @ B[Ktot,256])
__global__ void k_wmma_gemm(const float* A0, const float* A1, const float* A2,
                            const float* B, float* C, int M, int Ktot,
                            int applySigmoid) {
  int wave = threadIdx.x >> 5;
  int lane = threadIdx.x & 31;
  int row0 = blockIdx.x * 16;
  int col0 = (blockIdx.y * 4 + wave) * 16;
  bool laneLow = lane < 16;
  int r = row0 + (lane & 15);
  int cb = col0 + (lane & 15);

  v8f c = {0.f, 0.f, 0.f, 0.f, 0.f, 0.f, 0.f, 0.f};

#pragma unroll 4
  for (int k0 = 0; k0 < Ktot; k0 += 4) {
    const float* As = (k0 < 256) ? A0 : ((k0 < 512) ? A1 : A2);
    int kk = (k0 & 255) + (laneLow ? 0 : 2);
    v2f a;
    a.x = As[r * 256 + kk];
    a.y = As[r * 256 + kk + 1];
    int kb = k0 + (laneLow ? 0 : 2);
    v2f b;
    b.x = B[kb * 256 + cb];
    b.y = B[(kb + 1) * 256 + cb];
    c = __builtin_amdgcn_wmma_f32_16x16x4_f32(false, a, false, b, (short)0, c,
                                              false, false);
  }

#pragma unroll
  for (int i = 0; i < 8; ++i) {
    int rr = row0 + i + (laneLow ? 0 : 8);
    float v = c[i];
    if (applySigmoid) v = sigmoidf(v);
    C[rr * 256 + cb] = v;
  }
}

// ------------------------------------------------------ CSR construction
__global__ void k_count(const int* dst, int E, int* cnt) {
  int e = blockIdx.x * 256 + threadIdx.x;
  if (e < E) atomicAdd(cnt + dst[e], 1);
}

// single-block chunked exclusive scan: rowstart[0..n], cur = rowstart copy
__global__ void k_scan(const int* cnt, int* rowstart, int* cur, int n) {
  __shared__ int chunk[256];
  int t = threadIdx.x;
  int per = (n + 255) / 256;
  int b0 = t * per;
  int b1 = b0 + per < n ? b0 + per : n;
  int sum = 0;
  for (int i = b0; i < b1; ++i) sum += cnt[i];
  chunk[t] = sum;
  __syncthreads();
  if (t == 0) {
    int run = 0;
    for (int i = 0; i < 256; ++i) { int c = chunk[i]; chunk[i] = run; run += c; }
    rowstart[n] = run;
  }
  __syncthreads();
  int run = chunk[t];
  for (int i = b0; i < b1; ++i) {
    rowstart[i] = run; cur[i] = run;
    run += cnt[i];
  }
}

__global__ void k_scatter(const int* src, const int* dst, int E,
                          int* cur, int* csr_src) {
  int e = blockIdx.x * 256 + threadIdx.x;
  if (e >= E) return;
  int p = atomicAdd(cur + dst[e], 1);
  csr_src[p] = src[e];
}

// -------------------------------------- per-destination GAT (no atomics)
// out[d] = sigmoid( sum_j softmax_j(lrelu(ls[src_j]+ld[d])) * u[src_j] )
// Aggregation is chunked: per-edge weights computed ONCE (edge-parallel)
// into LDS, then the dim-parallel accumulation reads them via ds_load.
__global__ void k_gat_csr(const int* rowstart, const int* csr_src,
                          const float* ls, const float* ld,
                          const float* u, float* out) {
  __shared__ float red[256];
  __shared__ float att_sm[256];
  __shared__ int src_sm[256];
  int d = blockIdx.x, t = threadIdx.x;
  int beg = rowstart[d], end = rowstart[d + 1];
  float ldv = ld[d];

  float mx = -INFINITY;
  for (int j = beg + t; j < end; j += 256)
    mx = fmaxf(mx, lrelu(ls[csr_src[j]] + ldv));
  mx = blockReduceMax(mx, red);

  float sm = 0.f;
  for (int j = beg + t; j < end; j += 256)
    sm += __expf(lrelu(ls[csr_src[j]] + ldv) - mx);
  sm = blockReduceSum(sm, red);
  float inv = 1.f / sm;

  float acc = 0.f;
  for (int chunk = beg; chunk < end; chunk += 256) {
    int j = chunk + t;
    if (j < end) {
      int s = csr_src[j];
      src_sm[t] = s;
      att_sm[t] = __expf(lrelu(ls[s] + ldv) - mx) * inv;
    }
    __syncthreads();
    int n = end - chunk < 256 ? end - chunk : 256;
    for (int jj = 0; jj < n; ++jj)
      acc += att_sm[jj] * u[src_sm[jj] * 256 + t];
    __syncthreads();
  }
  out[d * 256 + t] = sigmoidf(acc);
}

// ------------------------------------------- supernode GAT (dst == 0)
__global__ void k_edge_logit0(const int* src, int E, const float* ls,
                              const float* ld0, float* elog, unsigned* menc) {
  int e = blockIdx.x * 256 + threadIdx.x;
  if (e >= E) return;
  float x = lrelu(ls[src[e]] + ld0[0]);
  elog[e] = x;
  unsigned u = __float_as_uint(x);
  u = (u & 0x80000000u) ? ~u : (u | 0x80000000u);
  atomicMax(menc, u);
}

__global__ void k_edge_exp0(int E, float* elog, const unsigned* menc, float* ssum) {
  int e = blockIdx.x * 256 + threadIdx.x;
  if (e >= E) return;
  unsigned u = menc[0];
  unsigned bits = (u & 0x80000000u) ? (u & 0x7FFFFFFFu) : ~u;
  float v = __expf(elog[e] - __uint_as_float(bits));
  elog[e] = v;
  atomicAdd(ssum, v);
}

// 128 edges per block accumulated in registers, 1 atomicAdd per dim per block
__global__ void k_super_agg(const int* src, int E, const float* elog,
                            const float* ssum, const float* u, float* agg) {
  int t = threadIdx.x;
  int base = blockIdx.x * 128;
  int lim = base + 128 < E ? base + 128 : E;
  float inv = 1.f / ssum[0];
  float acc = 0.f;
  for (int j = base; j < lim; ++j)
    acc += elog[j] * inv * u[src[j] * 256 + t];
  atomicAdd(agg + t, acc);
}

// ---------------------------------------------------------- small finals
__global__ void k_sup_out(const float* ns_sup, const float* nw_sup,
                          const float* super_h, const float* W, float* out) {
  int t = threadIdx.x;
  float acc = 0.f;
  for (int k = 0; k < 256; ++k) acc += ns_sup[k] * W[k * 256 + t];
  for (int k = 0; k < 256; ++k) acc += nw_sup[k] * W[(256 + k) * 256 + t];
  for (int k = 0; k < 256; ++k) acc += super_h[k] * W[(512 + k) * 256 + t];
  out[t] = sigmoidf(acc);
}

__global__ void k_scores(const float* super_h, const float* dots, float* out) {
  __shared__ float red[256];
  int t = threadIdx.x;
  float a = super_h[t];
  float n2 = blockReduceSum(a * a, red);
  float inv = rsqrtf(n2);
  float v0 = dots[t] * inv, v1 = dots[256 + t] * inv;
  float mx = blockReduceMax(fmaxf(v0, v1), red);
  float e0 = __expf(v0 - mx), e1 = __expf(v1 - mx);
  float sm = blockReduceSum(e0 + e1, red);
  out[t] = e0 / sm;
  out[256 + t] = e1 / sm;
}

// ---------------------------------------------------------------- launch

extern "C" void kernel_launch(void* const* d_in, const int* in_sizes, int n_in,
                              void* d_out, int out_size, void* d_ws, size_t ws_size,
                              hipStream_t stream) {
  const float* word_h  = (const float*)d_in[0];
  const float* sent_h  = (const float*)d_in[1];
  const float* super_h = (const float*)d_in[2];
  const float* score   = (const float*)d_in[3];
  const float* W_w     = (const float*)d_in[4];
  const float* W_s     = (const float*)d_in[5];
  const float* W_sup   = (const float*)d_in[6];
  const float* W_s_att = (const float*)d_in[7];
  const float* W_w_att = (const float*)d_in[8];
  const float* a_sw    = (const float*)d_in[9];
  const float* W_wh    = (const float*)d_in[10];
  const float* a_ss    = (const float*)d_in[11];
  const float* a_ws    = (const float*)d_in[12];
  const float* W_sh    = (const float*)d_in[13];
  const float* a_ssup  = (const float*)d_in[14];
  const float* a_wsup  = (const float*)d_in[15];
  const float* W_suph  = (const float*)d_in[16];
  const int* sw_src    = (const int*)d_in[17];
  const int* sw_dst    = (const int*)d_in[18];
  const int* ss_src    = (const int*)d_in[19];
  const int* ss_dst    = (const int*)d_in[20];
  const int* ws_src    = (const int*)d_in[21];
  const int* ws_dst    = (const int*)d_in[22];
  const int* ssup_src  = (const int*)d_in[23];
  const int* wsup_src  = (const int*)d_in[24];
  (void)in_sizes; (void)n_in; (void)out_size; (void)ws_size;

  float* out = (float*)d_out;
  float* o_scores = out;                          // 512
  float* o_s_out  = out + 512;                    // 512*256
  float* o_sup    = out + 512 + NS * 256;         // 256
  float* o_w_out  = out + 512 + NS * 256 + 256;   // 20000*256

  // ---- workspace bump allocator
  float* Wf = (float*)d_ws;
  size_t off = 0;
  auto alloc = [&](size_t n) { float* p = Wf + off; off += n; return p; };

  float* sh      = alloc((size_t)NS * 256);
  float* uw      = alloc((size_t)NW * 256);
  float* us      = alloc((size_t)NS * 256);
  float* proj    = alloc(10 * 256);
  float* ld_sw   = alloc(NW);
  float* ls_ws   = alloc(NW);
  float* ls_wsup = alloc(NW);
  float* ls_sw   = alloc(NS);
  float* ls_ss   = alloc(NS);
  float* ld_ss   = alloc(NS);
  float* ld_ws   = alloc(NS);
  float* ls_ssup = alloc(NS);
  float* dots    = alloc(NS);
  float* ld_ssup = alloc(1);
  float* ld_wsup = alloc(1);
  // CSR storage (ints)
  int* cnt_sw   = (int*)alloc(NW);
  int* row_sw   = (int*)alloc(NW + 1);
  int* cur_sw   = (int*)alloc(NW);
  int* csr_sw   = (int*)alloc(E_SW);
  int* cnt_ss   = (int*)alloc(NS);
  int* row_ss   = (int*)alloc(NS + 1);
  int* cur_ss   = (int*)alloc(NS);
  int* csr_ss   = (int*)alloc(E_SS);
  int* cnt_ws   = (int*)alloc(NS);
  int* row_ws   = (int*)alloc(NS + 1);
  int* cur_ws   = (int*)alloc(NS);
  int* csr_ws   = (int*)alloc(E_WS);
  // supernode GAT state
  float* e_ssup  = alloc(NS);
  float* e_wsup  = alloc(NW);
  unsigned* m_ssup = (unsigned*)alloc(1);
  unsigned* m_wsup = (unsigned*)alloc(1);
  float* s_ssup  = alloc(1);
  float* s_wsup  = alloc(1);
  // GAT outputs
  float* n_w     = alloc((size_t)NW * 256);
  float* ns_s    = alloc((size_t)NS * 256);
  float* nw_s    = alloc((size_t)NS * 256);
  float* ns_sup  = alloc(256);
  float* nw_sup  = alloc(256);

  auto blk = [](int n) { return (n + 255) / 256; };

  // ---- init: CSR counters + supernode softmax state
  k_fill_u32<<<blk(NW), 256, 0, stream>>>((unsigned*)cnt_sw, 0u, NW);
  k_fill_u32<<<blk(NS), 256, 0, stream>>>((unsigned*)cnt_ss, 0u, NS);
  k_fill_u32<<<blk(NS), 256, 0, stream>>>((unsigned*)cnt_ws, 0u, NS);
  k_fill_u32<<<1, 256, 0, stream>>>(m_ssup, ENC_NEG_INF, 1);
  k_fill_u32<<<1, 256, 0, stream>>>(m_wsup, ENC_NEG_INF, 1);
  k_fill_f32<<<1, 256, 0, stream>>>(s_ssup, 0.f, 1);
  k_fill_f32<<<1, 256, 0, stream>>>(s_wsup, 0.f, 1);
  k_fill_f32<<<1, 256, 0, stream>>>(ns_sup, 0.f, 256);
  k_fill_f32<<<1, 256, 0, stream>>>(nw_sup, 0.f, 256);

  // ---- projections and per-node attention logits
  k_proj<<<10, 256, 0, stream>>>(W_s, W_w, W_sup, a_sw, a_ss, a_ws,
                                 a_ssup, a_wsup, proj);
  k_sent<<<NS, 256, 0, stream>>>(sent_h, score, super_h, proj, sh,
                                 ls_sw, ls_ss, ld_ss, ld_ws, ls_ssup, dots);
  k_word<<<NW, 256, 0, stream>>>(word_h, proj, ld_sw, ls_ws, ls_wsup);
  k_sup_scalars<<<1, 256, 0, stream>>>(super_h, proj, ld_ssup, ld_wsup);

  // ---- attention-payload GEMMs (WMMA f32 16x16x4)
  k_wmma_gemm<<<dim3(NW / 16, 4), 128, 0, stream>>>(word_h, nullptr, nullptr,
                                                    W_w_att, uw, NW, 256, 0);
  k_wmma_gemm<<<dim3(NS / 16, 4), 128, 0, stream>>>(sh, nullptr, nullptr,
                                                    W_s_att, us, NS, 256, 0);

  // ---- CSR builds
  k_count<<<blk(E_SW), 256, 0, stream>>>(sw_dst, E_SW, cnt_sw);
  k_scan<<<1, 256, 0, stream>>>(cnt_sw, row_sw, cur_sw, NW);
  k_scatter<<<blk(E_SW), 256, 0, stream>>>(sw_src, sw_dst, E_SW, cur_sw, csr_sw);

  k_count<<<blk(E_SS), 256, 0, stream>>>(ss_dst, E_SS, cnt_ss);
  k_scan<<<1, 256, 0, stream>>>(cnt_ss, row_ss, cur_ss, NS);
  k_scatter<<<blk(E_SS), 256, 0, stream>>>(ss_src, ss_dst, E_SS, cur_ss, csr_ss);

  k_count<<<blk(E_WS), 256, 0, stream>>>(ws_dst, E_WS, cnt_ws);
  k_scan<<<1, 256, 0, stream>>>(cnt_ws, row_ws, cur_ws, NS);
  k_scatter<<<blk(E_WS), 256, 0, stream>>>(ws_src, ws_dst, E_WS, cur_ws, csr_ws);

  // ---- main GATs: per-dst, atomic-free, sigmoid fused
  k_gat_csr<<<NW, 256, 0, stream>>>(row_sw, csr_sw, ls_sw, ld_sw, us, n_w);
  k_gat_csr<<<NS, 256, 0, stream>>>(row_ss, csr_ss, ls_ss, ld_ss, us, ns_s);
  k_gat_csr<<<NS, 256, 0, stream>>>(row_ws, csr_ws, ls_ws, ld_ws, uw, nw_s);

  // ---- supernode GATs (all edges -> node 0)
  k_edge_logit0<<<blk(NS), 256, 0, stream>>>(ssup_src, NS, ls_ssup, ld_ssup, e_ssup, m_ssup);
  k_edge_exp0<<<blk(NS), 256, 0, stream>>>(NS, e_ssup, m_ssup, s_ssup);
  k_super_agg<<<(NS + 127) / 128, 256, 0, stream>>>(ssup_src, NS, e_ssup, s_ssup, us, ns_sup);

  k_edge_logit0<<<blk(NW), 256, 0, stream>>>(wsup_src, NW, ls_wsup, ld_wsup, e_wsup, m_wsup);
  k_edge_exp0<<<blk(NW), 256, 0, stream>>>(NW, e_wsup, m_wsup, s_wsup);
  k_super_agg<<<(NW + 127) / 128, 256, 0, stream>>>(wsup_src, NW, e_wsup, s_wsup, uw, nw_sup);

  k_sigmoid<<<1, 256, 0, stream>>>(ns_sup, 256);
  k_sigmoid<<<1, 256, 0, stream>>>(nw_sup, 256);

  // ---- output GEMMs (fused sigmoid)
  k_wmma_gemm<<<dim3(NW / 16, 4), 128, 0, stream>>>(word_h, n_w, nullptr,
                                                    W_wh, o_w_out, NW, 512, 1);
  k_wmma_gemm<<<dim3(NS / 16, 4), 128, 0, stream>>>(ns_s, nw_s, sh,
                                                    W_sh, o_s_out, NS, 768, 1);
  k_sup_out<<<1, 256, 0, stream>>>(ns_sup, nw_sup, super_h, W_suph, o_sup);
  k_scores<<<1, 256, 0, stream>>>(super_h, dots, o_scores);
}